// SM2_44985487458427
// MI455X (gfx1250) — compile-verified
//
#include <hip/hip_runtime.h>
#include <stdint.h>

// SM-2 spaced-repetition scan, B independent rows x 400 sequential steps.
// Memory-bound-ish (839 MB in / 23.3 TB/s ~= 36 us floor) AND VALU-heavy
// (~3.1 G lane-ops). Strategy: one thread per row; block-cooperative
// 256x16 tile staging via CDNA5 async global->LDS copies, double buffered
// so the HBM stream overlaps the VALU recurrence.

#define ROWS_PER_BLOCK 256
#define COLS_PER_CHUNK 16
#define NSTEPS         400
#define NCHUNK         (NSTEPS / COLS_PER_CHUNK)  // 25
#define LDS_STRIDE     20  // floats per LDS row: 80B pitch, 16B-aligned for b128,
                           // and 20*t mod 64 banks -> only 2-way conflict on reads.

__launch_bounds__(ROWS_PER_BLOCK)
__global__ void sm2_scan_async(const float* __restrict__ hist,
                               float* __restrict__ out, int B) {
  __shared__ float tile[2][ROWS_PER_BLOCK * LDS_STRIDE];

  const int t      = threadIdx.x;
  const int r0     = blockIdx.x * ROWS_PER_BLOCK;
  const int my_row = r0 + t;

  // Tile loader mapping: flat float4 index f = t + k*256, row = f>>2, seg = f&3.
  const int seg      = t & 3;   // which float4 inside the 16-col chunk
  const int row_base = t >> 2;  // 0..63

  // Low 32 bits of a flat pointer to __shared__ == workgroup-relative LDS byte
  // address (LDS aperture keeps offset in addr[31:0]).
  const unsigned lds0        = (unsigned)(uintptr_t)&tile[0][0];
  const unsigned row_pitch_b = LDS_STRIDE * 4u;
  const unsigned buf_bytes   = ROWS_PER_BLOCK * LDS_STRIDE * 4u;

  auto issue_chunk = [&](int chunk, int buf) {
#pragma unroll
    for (int k = 0; k < 4; ++k) {
      int row  = row_base + k * 64;   // 0..255
      int grow = r0 + row;
      if (grow >= B) grow = B - 1;    // clamp (harmless duplicate load)
      unsigned voff = (unsigned)grow * (NSTEPS * 4u)
                    + (unsigned)chunk * (COLS_PER_CHUNK * 4u)
                    + (unsigned)seg * 16u;
      unsigned laddr = lds0 + (unsigned)buf * buf_bytes
                     + (unsigned)row * row_pitch_b + (unsigned)seg * 16u;
      // CDNA5 async copy: 16B global -> LDS, tracked with ASYNCcnt.
      asm volatile("global_load_async_to_lds_b128 %0, %1, %2"
                   :
                   : "v"(laddr), "v"(voff), "s"(hist)
                   : "memory");
    }
  };

  issue_chunk(0, 0);

  float I  = 1.0f;
  float EF = 2.5f;
  int   n  = 0;  // saturating repetition counter: 0, 1, 2(==">=2")

  for (int c = 0; c < NCHUNK; ++c) {
    if (c + 1 < NCHUNK) {
      // Prefetch next chunk into the other buffer, then wait until only those
      // 4 outstanding async ops remain => chunk c's loads have landed.
      issue_chunk(c + 1, (c + 1) & 1);
      asm volatile("s_wait_asynccnt 4" ::: "memory");
    } else {
      asm volatile("s_wait_asynccnt 0" ::: "memory");
    }
    __syncthreads();  // all waves' async loads for chunk c are now visible

    const float* p = &tile[c & 1][t * LDS_STRIDE];
#pragma unroll
    for (int i = 0; i < COLS_PER_CHUNK; ++i) {
      float pv      = p[i];
      float q       = pv * 5.0f;
      bool  correct = (q >= 3.0f);
      bool  brk     = (pv == -1.0f);

      float In = (n >= 2) ? (I * EF) : ((n == 1) ? 6.0f : 1.0f);
      In = correct ? In : 1.0f;
      I  = brk ? I : In;
      I  = fminf(fmaxf(I, 1.0f), 274.0f);

      float u   = 5.0f - q;
      float EFn = EF + (0.1f - u * __builtin_fmaf(u, 0.02f, 0.08f));
      EFn = fmaxf(EFn, 1.3f);
      EF  = correct ? EFn : EF;

      n = correct ? ((n < 2) ? (n + 1) : 2) : 0;
    }
    __syncthreads();  // buffer (c&1) free for reuse at chunk c+2
  }

  if (my_row < B)
    out[my_row] = fminf(fmaxf(I, 1.0e-4f), 274.0f);
}

extern "C" void kernel_launch(void* const* d_in, const int* in_sizes, int n_in,
                              void* d_out, int out_size, void* d_ws, size_t ws_size,
                              hipStream_t stream) {
  (void)in_sizes; (void)n_in; (void)d_ws; (void)ws_size;
  const float* hist = (const float*)d_in[0];
  float*       out  = (float*)d_out;
  const int    B    = out_size;  // 524288 rows; in_sizes[0] == B * 400
  const int    grid = (B + ROWS_PER_BLOCK - 1) / ROWS_PER_BLOCK;
  sm2_scan_async<<<grid, ROWS_PER_BLOCK, 0, stream>>>(hist, out, B);
}